// DEQ_24979529793554
// MI455X (gfx1250) — compile-verified
//
#include <hip/hip_runtime.h>
#include <math.h>

typedef __attribute__((ext_vector_type(2))) float v2f;
typedef __attribute__((ext_vector_type(8))) float v8f;

#define MB   16384   // batch rows (M)
#define SS   512     // n_states
#define NIN  64
#define NOUT 64
#define EPSV 1e-3f
#define N_FIXED_ITERS 64   // total applications of f (1 explicit + 63 GEMM iters)

// ---------------------------------------------------------------------------
// helpers
// ---------------------------------------------------------------------------
static __device__ __forceinline__ float wave_reduce_add(float s) {
#pragma unroll
  for (int off = 16; off > 0; off >>= 1) s += __shfl_xor(s, off, 32);
  return s;
}

// ---------------------------------------------------------------------------
// Generic 128x64-tile GEMM on V_WMMA_F32_16X16X4_F32 (wave32, f32 in/out).
//   C[m,n] = sum_k A'[m,k] * B'[k,n]
//   btrans=1 : B'[k,n] = Bsrc[n*ldb + k]   (i.e. multiply by Bsrc^T)
//   btrans=0 : B'[k,n] = Bsrc[k*ldb + n]
// MODE 0: D = C + bias[n]                         (xA and y kernels)
// MODE 1: D = tanh(C + Cadd[m,n]); optional residual vs Asrc + converged flag
// MODE 2: A'[m,k] = Asrc * (1 - Aux^2); accumulate sum(C^2) into accums[0]
// ---------------------------------------------------------------------------
template <int MODE>
__global__ void __launch_bounds__(256)
deq_gemm(const float* __restrict__ Asrc, int lda,
         const float* __restrict__ Aux,
         const float* __restrict__ Bsrc, int ldb, int btrans,
         int K,
         const float* __restrict__ Cadd,
         float* __restrict__ Dout, int ldd,
         float* __restrict__ accums,
         const int* __restrict__ flagp,
         int respect_flag, int do_resid) {
  const int tid   = threadIdx.x;
  const int lane  = tid & 31;
  const int wv    = tid >> 5;
  const int m_blk = blockIdx.x * 128;
  const int n_blk = blockIdx.y * 64;
  const int r16   = lane & 15;
  const int hi    = (lane < 16) ? 0 : 1;

  if (MODE == 1 && respect_flag) {
    if (*(volatile const int*)flagp) {
      // Converged: propagate z_star unchanged through the ping-pong buffer.
#pragma unroll
      for (int t = 0; t < 4; ++t) {
        const int n = n_blk + t * 16 + r16;
#pragma unroll
        for (int i = 0; i < 8; ++i) {
          const size_t idx = (size_t)(m_blk + wv * 16 + hi * 8 + i) * SS + n;
          Dout[idx] = Asrc[idx];
        }
      }
      return;
    }
  }

  // Row pad 64 -> 68 floats: bank = (row*68 + k) % 64 = (4*row + k) % 64,
  // conflict-free fragment reads; 272-byte rows keep float4 stores aligned.
  __shared__ float ldsA[128 * 68];
  __shared__ float ldsB[64 * 68];

  v8f acc[4];
#pragma unroll
  for (int t = 0; t < 4; ++t)
#pragma unroll
    for (int j = 0; j < 8; ++j) acc[t][j] = 0.f;

  const int sel = hi * 2;  // K-pair selector for 32-bit WMMA fragments

  for (int kc = 0; kc < K; kc += 64) {
    // ---- stage A panel: 128 rows x 64 k ----
#pragma unroll
    for (int ii = 0; ii < 8; ++ii) {
      const int i  = tid + ii * 256;
      const int r  = i >> 4;
      const int c4 = (i & 15) << 2;
      const size_t goff = (size_t)(m_blk + r) * lda + kc + c4;
      float4 val = *(const float4*)(Asrc + goff);
      if (MODE == 2) {
        float4 w4 = *(const float4*)(Aux + goff);
        val.x *= 1.f - w4.x * w4.x;
        val.y *= 1.f - w4.y * w4.y;
        val.z *= 1.f - w4.z * w4.z;
        val.w *= 1.f - w4.w * w4.w;
      }
      *(float4*)(&ldsA[r * 68 + c4]) = val;
    }
    // ---- stage B panel: 64 cols (n) x 64 k, stored [n][k] ----
    if (btrans) {
#pragma unroll
      for (int ii = 0; ii < 4; ++ii) {
        const int i  = tid + ii * 256;
        const int r  = i >> 4;
        const int c4 = (i & 15) << 2;
        float4 val = *(const float4*)(Bsrc + (size_t)(n_blk + r) * ldb + kc + c4);
        *(float4*)(&ldsB[r * 68 + c4]) = val;
      }
    } else {
#pragma unroll
      for (int ii = 0; ii < 16; ++ii) {
        const int i = tid + ii * 256;
        const int k = i >> 6;
        const int n = i & 63;
        ldsB[n * 68 + k] = Bsrc[(size_t)(kc + k) * ldb + n_blk + n];
      }
    }
    __syncthreads();

    // ---- 16 K-steps of V_WMMA_F32_16X16X4_F32, 4 N-tiles per wave ----
    const float* pa = &ldsA[(wv * 16 + r16) * 68 + sel];
    const float* pb = &ldsB[r16 * 68 + sel];
#pragma unroll
    for (int kk = 0; kk < 64; kk += 4) {
      v2f a = *(const v2f*)(pa + kk);
#pragma unroll
      for (int t = 0; t < 4; ++t) {
        v2f b = *(const v2f*)(pb + t * 16 * 68 + kk);
        acc[t] = __builtin_amdgcn_wmma_f32_16x16x4_f32(
            false, a, false, b, (short)0, acc[t], false, false);
      }
    }
    __syncthreads();
  }

  // ---- epilogue ----
  const int mrow0 = m_blk + wv * 16 + hi * 8;
  if (MODE == 0) {
#pragma unroll
    for (int t = 0; t < 4; ++t) {
      const int n = n_blk + t * 16 + r16;
      const float bias = Cadd[n];
#pragma unroll
      for (int i = 0; i < 8; ++i)
        Dout[(size_t)(mrow0 + i) * ldd + n] = acc[t][i] + bias;
    }
  } else if (MODE == 1) {
    float s_num = 0.f, s_den = 0.f;
#pragma unroll
    for (int t = 0; t < 4; ++t) {
      const int n = n_blk + t * 16 + r16;
#pragma unroll
      for (int i = 0; i < 8; ++i) {
        const size_t idx = (size_t)(mrow0 + i) * SS + n;
        const float val = tanhf(acc[t][i] + Cadd[idx]);
        Dout[idx] = val;
        if (do_resid) {
          const float d = val - Asrc[idx];
          s_num += d * d;
          s_den += val * val;
        }
      }
    }
    if (do_resid) {
      s_num = wave_reduce_add(s_num);
      s_den = wave_reduce_add(s_den);
      if (lane == 0) {
        atomicAdd(&accums[0], s_num);
        atomicAdd(&accums[1], s_den);
      }
    }
  } else {  // MODE 2: Hutchinson sum of squares
    float s = 0.f;
#pragma unroll
    for (int t = 0; t < 4; ++t)
#pragma unroll
      for (int i = 0; i < 8; ++i) s += acc[t][i] * acc[t][i];
    s = wave_reduce_add(s);
    if (lane == 0) atomicAdd(&accums[0], s);
  }
}

// ---------------------------------------------------------------------------
__global__ void deq_init(float* accums, int* flag) {
  accums[0] = 0.f;
  accums[1] = 0.f;
  accums[2] = 0.f;
  *flag = 0;
}

// z1 = tanh(xA)   (first application of f at z0 = 0; b already folded into xA)
__global__ void __launch_bounds__(256)
deq_tanh_init(const float* __restrict__ xA, float* __restrict__ z) {
  size_t i = (size_t)blockIdx.x * blockDim.x + threadIdx.x;
  const size_t stride = (size_t)gridDim.x * blockDim.x;
  const size_t n4 = (size_t)MB * SS / 4;
  const float4* src = (const float4*)xA;
  float4* dst = (float4*)z;
  for (; i < n4; i += stride) {
    float4 u = src[i];
    float4 o;
    o.x = tanhf(u.x);
    o.y = tanhf(u.y);
    o.z = tanhf(u.z);
    o.w = tanhf(u.w);
    dst[i] = o;
  }
}

// Relative-residual stopping rule (matches reference: stop when res <= eps)
__global__ void deq_rescheck(float* accums, int* flag) {
  if (*flag == 0) {
    const float res = sqrtf(accums[0]) / (sqrtf(accums[1]) + 1e-8f);
    if (res <= EPSV) *flag = 1;
  }
  accums[0] = 0.f;
  accums[1] = 0.f;
}

__global__ void deq_finish(const float* accums, float* outp) {
  *outp = accums[2] / ((float)MB * (float)SS);
}

// ---------------------------------------------------------------------------
extern "C" void kernel_launch(void* const* d_in, const int* in_sizes, int n_in,
                              void* d_out, int out_size, void* d_ws, size_t ws_size,
                              hipStream_t stream) {
  const float* x  = (const float*)d_in[0];  // (16384, 64)
  const float* Am = (const float*)d_in[1];  // (512, 64)
  const float* Bm = (const float*)d_in[2];  // (512, 512)
  const float* bv = (const float*)d_in[3];  // (512,)
  const float* Wh = (const float*)d_in[4];  // (64, 512)
  const float* bh = (const float*)d_in[5];  // (64,)
  const float* v  = (const float*)d_in[6];  // (16384, 512)
  float* out = (float*)d_out;               // y (16384*64) then jac_loss (1)

  char* ws = (char*)d_ws;
  const size_t matB = (size_t)MB * SS * sizeof(float);  // 32 MB
  float* xA  = (float*)(ws);
  float* zA  = (float*)(ws + matB);
  float* zB  = (float*)(ws + 2 * matB);
  float* acc = (float*)(ws + 3 * matB);  // [0]=res_num [1]=res_den [2]=jac
  int*  flag = (int*)(acc + 3);

  const dim3 blk(256);
  deq_init<<<1, 1, 0, stream>>>(acc, flag);

  // xA = x @ A^T + b   (bias folded in so the iteration only adds xA)
  deq_gemm<0><<<dim3(128, 8), blk, 0, stream>>>(x, NIN, nullptr, Am, NIN, 1, NIN,
                                                bv, xA, SS, nullptr, nullptr, 0, 0);
  // z1 = tanh(xA)  (res0 = 1 > eps, so no check needed)
  deq_tanh_init<<<dim3(2048), blk, 0, stream>>>(xA, zA);

  float* cur = zA;
  float* nxt = zB;
  for (int it = 0; it < N_FIXED_ITERS - 1; ++it) {
    deq_gemm<1><<<dim3(128, 8), blk, 0, stream>>>(cur, SS, nullptr, Bm, SS, 1, SS,
                                                  xA, nxt, SS, acc, flag, 1, 1);
    deq_rescheck<<<1, 1, 0, stream>>>(acc, flag);
    float* t = cur; cur = nxt; nxt = t;
  }
  // cur == z_star. Reference's VJP uses w = f(z_star): force one more apply.
  deq_gemm<1><<<dim3(128, 8), blk, 0, stream>>>(cur, SS, nullptr, Bm, SS, 1, SS,
                                                xA, nxt, SS, nullptr, nullptr, 0, 0);
  // y = z_star @ Wh^T + bh
  deq_gemm<0><<<dim3(128, 1), blk, 0, stream>>>(cur, SS, nullptr, Wh, SS, 1, SS,
                                                bh, out, NOUT, nullptr, nullptr, 0, 0);
  // jac_acc = sum( ((v .* (1 - w^2)) @ Bm)^2 )
  deq_gemm<2><<<dim3(128, 8), blk, 0, stream>>>(v, SS, nxt, Bm, SS, 0, SS,
                                                nullptr, nullptr, 0, acc + 2,
                                                nullptr, 0, 0);
  deq_finish<<<1, 1, 0, stream>>>(acc, out + (size_t)MB * NOUT);
}